// Hierarchical_9DoF_Registration_86792699118036
// MI455X (gfx1250) — compile-verified
//
#include <hip/hip_runtime.h>

typedef float v2f __attribute__((ext_vector_type(2)));
typedef float v8f __attribute__((ext_vector_type(8)));

#define NEG_SLOPE 0.2f
#define VN_EPS 1e-6f

// ---------------------------------------------------------------------------
// Phase 1: streaming mean over the point dimension (bandwidth-bound, ~673 MB).
// One wave per row of N contiguous floats; float4 loads; result written
// directly in GEMM layout: dst[(chanOff+c)*192 + d*64 + b].
// ---------------------------------------------------------------------------
__global__ void mean_rows_kernel(const float* __restrict__ src,
                                 float* __restrict__ dst,
                                 int C, int N, int chanOff)
{
    const int lane = threadIdx.x & 31;
    const int waveInBlock = threadIdx.x >> 5;
    const int row = blockIdx.x * (blockDim.x >> 5) + waveInBlock;
    const int totalRows = 64 * C * 3;          // B * C * 3
    if (row >= totalRows) return;

    const float* p = src + (size_t)row * N;
    float s = 0.f;
    for (int i = lane * 4; i < N; i += 32 * 4) {
        float4 v = *reinterpret_cast<const float4*>(p + i);
        s += v.x + v.y + v.z + v.w;
    }
    #pragma unroll
    for (int off = 16; off > 0; off >>= 1)
        s += __shfl_xor(s, off, 32);

    if (lane == 0) {
        int d   = row % 3;
        int tmp = row / 3;
        int c   = tmp % C;
        int b   = tmp / C;
        dst[(size_t)(chanOff + c) * 192 + d * 64 + b] = s / (float)N;
    }
}

// ---------------------------------------------------------------------------
// Phase 2: P = W*X, D = Wdir*X as one 512x192x512 GEMM per side via
// V_WMMA_F32_16X16X4_F32, fused VN-LeakyReLU + mean-centering epilogue.
// Columns are d-major (n = d*64 + b): each wave owns rows mBase..+15 and
// batches bBase..+15 for ALL three d components (6 accumulators), so the
// 3-vector nonlinearity is per-lane.
//
// Fragment layouts (ISA 7.12.2, 32-bit, wave32):
//   A (16x4):  v0 = K=k0+2*hi,   v1 = K=k0+2*hi+1, row M = mBase + (lane&15)
//   B (4x16):  v0 = K=k0+2*hi,   v1 = K=k0+2*hi+1, col N = nBase + (lane&15)
//   C/D (16x16): vgpr r -> M = mBase + r + 8*hi, N = nBase + (lane&15)
// ---------------------------------------------------------------------------
__global__ void vn_mlp_wmma_kernel(const float* __restrict__ W,
                                   const float* __restrict__ Wdir,
                                   const float* __restrict__ Xmat, // (512,192)
                                   const float* __restrict__ Ymat, // (512,192)
                                   const float* __restrict__ mux,  // (64,3)
                                   const float* __restrict__ muy,  // (64,3)
                                   float* __restrict__ XC,         // (512,192)
                                   float* __restrict__ YC)         // (512,192)
{
    const float* Xm = blockIdx.z ? Ymat : Xmat;
    const float* mu = blockIdx.z ? muy  : mux;
    float*       out = blockIdx.z ? YC  : XC;

    const int lane  = threadIdx.x & 31;
    const int wave  = threadIdx.x >> 5;
    const int mTile = blockIdx.x * 4 + wave;   // 0..31
    const int bTile = blockIdx.y;              // 0..3
    const int mBase = mTile * 16;
    const int bBase = bTile * 16;

    const int lo = lane & 15;
    const int hi = lane >> 4;

    v8f accP0 = {}, accP1 = {}, accP2 = {};
    v8f accD0 = {}, accD1 = {}, accD2 = {};

    const float* wRow  = W    + (size_t)(mBase + lo) * 512;
    const float* wdRow = Wdir + (size_t)(mBase + lo) * 512;
    const int col = bBase + lo;

    for (int k0 = 0; k0 < 512; k0 += 4) {
        const int kk = k0 + 2 * hi;
        v2f aW  = *reinterpret_cast<const v2f*>(wRow  + kk);
        v2f aWd = *reinterpret_cast<const v2f*>(wdRow + kk);

        const float* xp = Xm + (size_t)kk * 192 + col;
        v2f b0, b1, b2;
        b0.x = xp[0];        b0.y = xp[192];          // d = 0
        b1.x = xp[64];       b1.y = xp[192 + 64];     // d = 1
        b2.x = xp[128];      b2.y = xp[192 + 128];    // d = 2

        accP0 = __builtin_amdgcn_wmma_f32_16x16x4_f32(false, aW,  false, b0, (short)0, accP0, false, false);
        accP1 = __builtin_amdgcn_wmma_f32_16x16x4_f32(false, aW,  false, b1, (short)0, accP1, false, false);
        accP2 = __builtin_amdgcn_wmma_f32_16x16x4_f32(false, aW,  false, b2, (short)0, accP2, false, false);
        accD0 = __builtin_amdgcn_wmma_f32_16x16x4_f32(false, aWd, false, b0, (short)0, accD0, false, false);
        accD1 = __builtin_amdgcn_wmma_f32_16x16x4_f32(false, aWd, false, b1, (short)0, accD1, false, false);
        accD2 = __builtin_amdgcn_wmma_f32_16x16x4_f32(false, aWd, false, b2, (short)0, accD2, false, false);
    }

    // Epilogue: VN-LeakyReLU + subtract mu (all per-lane)
    const int bIdx = col;                 // batch index of this lane's column
    const float mu0 = mu[bIdx * 3 + 0];
    const float mu1 = mu[bIdx * 3 + 1];
    const float mu2 = mu[bIdx * 3 + 2];

    #pragma unroll
    for (int r = 0; r < 8; ++r) {
        const int m = mBase + r + 8 * hi;
        float p0 = accP0[r], p1 = accP1[r], p2 = accP2[r];
        float d0 = accD0[r], d1 = accD1[r], d2 = accD2[r];
        float dot = p0 * d0 + p1 * d1 + p2 * d2;
        float dd  = d0 * d0 + d1 * d1 + d2 * d2;
        float f   = dot / (dd + VN_EPS);
        float q0 = p0 - f * d0, q1 = p1 - f * d1, q2 = p2 - f * d2;
        bool pos = (dot >= 0.f);
        float n0 = pos ? p0 : q0;
        float n1 = pos ? p1 : q1;
        float n2 = pos ? p2 : q2;
        float z0 = NEG_SLOPE * p0 + (1.f - NEG_SLOPE) * n0;
        float z1 = NEG_SLOPE * p1 + (1.f - NEG_SLOPE) * n1;
        float z2 = NEG_SLOPE * p2 + (1.f - NEG_SLOPE) * n2;
        out[(size_t)m * 192 +       bIdx] = z0 - mu0;
        out[(size_t)m * 192 +  64 + bIdx] = z1 - mu1;
        out[(size_t)m * 192 + 128 + bIdx] = z2 - mu2;
    }
}

// ---------------------------------------------------------------------------
// Phase 3: per-batch 3x3 covariance + norms (wave reduction), then
// R = U*Vt == orthogonal polar factor of cov via Newton X <- 0.5(X + X^-T),
// then t = mu_y - R mu_x and scale = |yc| / (|xc| + eps).
// Output: R[64*9] | t[64*3] | scale[64]
// ---------------------------------------------------------------------------
__global__ void kabsch_kernel(const float* __restrict__ XC,
                              const float* __restrict__ YC,
                              const float* __restrict__ mux,
                              const float* __restrict__ muy,
                              float* __restrict__ outp)
{
    const int b    = blockIdx.x;     // 0..63
    const int lane = threadIdx.x;    // 0..31

    float cov[9] = {0.f,0.f,0.f,0.f,0.f,0.f,0.f,0.f,0.f};
    float xn2 = 0.f, yn2 = 0.f;

    for (int m = lane; m < 512; m += 32) {
        float xv[3], yv[3];
        #pragma unroll
        for (int d = 0; d < 3; ++d) {
            xv[d] = XC[(size_t)m * 192 + d * 64 + b];
            yv[d] = YC[(size_t)m * 192 + d * 64 + b];
            xn2 += xv[d] * xv[d];
            yn2 += yv[d] * yv[d];
        }
        #pragma unroll
        for (int d = 0; d < 3; ++d)
            #pragma unroll
            for (int e = 0; e < 3; ++e)
                cov[d * 3 + e] += xv[d] * yv[e];
    }

    #pragma unroll
    for (int off = 16; off > 0; off >>= 1) {
        #pragma unroll
        for (int i = 0; i < 9; ++i) cov[i] += __shfl_xor(cov[i], off, 32);
        xn2 += __shfl_xor(xn2, off, 32);
        yn2 += __shfl_xor(yn2, off, 32);
    }

    if (lane != 0) return;

    // Newton polar iteration on normalized cov
    float X[9];
    float nrm = 0.f;
    #pragma unroll
    for (int i = 0; i < 9; ++i) nrm += cov[i] * cov[i];
    nrm = sqrtf(nrm) + VN_EPS;
    #pragma unroll
    for (int i = 0; i < 9; ++i) X[i] = cov[i] / nrm;

    for (int it = 0; it < 24; ++it) {
        // cofactor matrix / det  ==  inv(X)^T
        float c00 = X[4]*X[8] - X[5]*X[7];
        float c01 = X[5]*X[6] - X[3]*X[8];
        float c02 = X[3]*X[7] - X[4]*X[6];
        float c10 = X[2]*X[7] - X[1]*X[8];
        float c11 = X[0]*X[8] - X[2]*X[6];
        float c12 = X[1]*X[6] - X[0]*X[7];
        float c20 = X[1]*X[5] - X[2]*X[4];
        float c21 = X[2]*X[3] - X[0]*X[5];
        float c22 = X[0]*X[4] - X[1]*X[3];
        float det = X[0]*c00 + X[1]*c01 + X[2]*c02;
        float id  = 1.f / (det + (det >= 0.f ? VN_EPS : -VN_EPS));
        X[0] = 0.5f * (X[0] + c00 * id);
        X[1] = 0.5f * (X[1] + c01 * id);
        X[2] = 0.5f * (X[2] + c02 * id);
        X[3] = 0.5f * (X[3] + c10 * id);
        X[4] = 0.5f * (X[4] + c11 * id);
        X[5] = 0.5f * (X[5] + c12 * id);
        X[6] = 0.5f * (X[6] + c20 * id);
        X[7] = 0.5f * (X[7] + c21 * id);
        X[8] = 0.5f * (X[8] + c22 * id);
    }

    #pragma unroll
    for (int i = 0; i < 9; ++i) outp[b * 9 + i] = X[i];

    const float mx0 = mux[b*3+0], mx1 = mux[b*3+1], mx2 = mux[b*3+2];
    #pragma unroll
    for (int i = 0; i < 3; ++i)
        outp[576 + b * 3 + i] =
            muy[b*3+i] - (X[i*3+0]*mx0 + X[i*3+1]*mx1 + X[i*3+2]*mx2);

    outp[768 + b] = sqrtf(yn2) / (sqrtf(xn2) + VN_EPS);
}

// ---------------------------------------------------------------------------
extern "C" void kernel_launch(void* const* d_in, const int* in_sizes, int n_in,
                              void* d_out, int out_size, void* d_ws, size_t ws_size,
                              hipStream_t stream) {
    // setup_inputs order: x0,y0,x1,y1,x2,y2,x3,y3,mu_x,mu_y,W,W_dir
    const float* xs[4] = {(const float*)d_in[0], (const float*)d_in[2],
                          (const float*)d_in[4], (const float*)d_in[6]};
    const float* ys[4] = {(const float*)d_in[1], (const float*)d_in[3],
                          (const float*)d_in[5], (const float*)d_in[7]};
    const float* mux = (const float*)d_in[8];
    const float* muy = (const float*)d_in[9];
    const float* W   = (const float*)d_in[10];
    const float* Wd  = (const float*)d_in[11];

    float* ws   = (float*)d_ws;
    float* Xmat = ws;                  // 512*192
    float* Ymat = ws + 98304;
    float* XCm  = ws + 2 * 98304;
    float* YCm  = ws + 3 * 98304;

    const int chans[4] = {64, 128, 128, 192};
    const int npts[4]  = {4096, 1024, 256, 64};
    int off = 0;
    for (int i = 0; i < 4; ++i) {
        const int rows   = 64 * chans[i] * 3;
        const int blocks = (rows + 7) / 8;      // 8 waves per 256-thread block
        mean_rows_kernel<<<blocks, 256, 0, stream>>>(xs[i], Xmat, chans[i], npts[i], off);
        mean_rows_kernel<<<blocks, 256, 0, stream>>>(ys[i], Ymat, chans[i], npts[i], off);
        off += chans[i];
    }

    dim3 g2(8, 4, 2);   // 8 blocks x 4 waves = 32 M-tiles; 4 batch tiles; 2 sides
    vn_mlp_wmma_kernel<<<g2, 128, 0, stream>>>(W, Wd, Xmat, Ymat, mux, muy, XCm, YCm);

    kabsch_kernel<<<64, 32, 0, stream>>>(XCm, YCm, mux, muy, (float*)d_out);
}